// HiGCN_46076409151505
// MI455X (gfx1250) — compile-verified
//
#include <hip/hip_runtime.h>
#include <math.h>

// ---------------- problem constants (match reference) ----------------
#define N_NODES 50000
#define EDGES   800000
#define FDIM    128
#define HDIM    128
#define CDIM    10
#define LAYERS  3
#define ORDER   2
#define HOPS    10
#define GRAPHS  128

typedef __attribute__((ext_vector_type(2))) float v2f;
typedef __attribute__((ext_vector_type(8))) float v8f;

// ---------------------------------------------------------------------
// Zero-fill (n must be a multiple of 4)
__global__ void fill_zero_kernel(float* __restrict__ p, int n4) {
    int i = blockIdx.x * blockDim.x + threadIdx.x;
    if (i < n4) {
        float4 z; z.x = 0.f; z.y = 0.f; z.z = 0.f; z.w = 0.f;
        ((float4*)p)[i] = z;
    }
}

// ---------------------------------------------------------------------
// SpMM scatter: zout[dst] += w * zin[src], one wave (32 lanes) per edge,
// float4 per lane => fully coalesced 512B gather, f32 atomics stay in L2
// (both the 25.6MB source and target fit in the 192MB L2).
__global__ void spmm_atomic_kernel(const int* __restrict__ dst,
                                   const int* __restrict__ src,
                                   const float* __restrict__ w,
                                   const float* __restrict__ zin,
                                   float* __restrict__ zout, int E) {
    int gid  = blockIdx.x * blockDim.x + threadIdx.x;
    int e    = gid >> 5;        // wave-uniform edge id
    int lane = gid & 31;
    if (e >= E) return;
    int   d  = dst[e];
    int   s  = src[e];
    float wt = w[e];
    float4 zv = ((const float4*)(zin + (size_t)s * HDIM))[lane];
    float* zo = zout + (size_t)d * HDIM + lane * 4;
    atomicAdd(zo + 0, wt * zv.x);
    atomicAdd(zo + 1, wt * zv.y);
    atomicAdd(zo + 2, wt * zv.z);
    atomicAdd(zo + 3, wt * zv.w);
}

// ---------------------------------------------------------------------
// Strided axpy: dst[r*ldd + 4f..] += coef * z[r*HDIM + 4f..]
// coef is broadcast-loaded from device memory (fW lives on device).
__global__ void axpy_strided_kernel(float* __restrict__ dst, int ldd,
                                    const float* __restrict__ z,
                                    const float* __restrict__ coefp, int rows) {
    float c = coefp[0];
    int tid = blockIdx.x * blockDim.x + threadIdx.x;
    int r = tid >> 5, fc = tid & 31;
    if (r >= rows) return;
    float4 zv = ((const float4*)(z + (size_t)r * HDIM))[fc];
    float4* dp = (float4*)(dst + (size_t)r * ldd) + fc;
    float4 dv = *dp;
    dv.x += c * zv.x; dv.y += c * zv.y; dv.z += c * zv.z; dv.w += c * zv.w;
    *dp = dv;
}

// ---------------------------------------------------------------------
// global_add_pool: pool[batch[n]] += h[n]
__global__ void pool_kernel(const float* __restrict__ h,
                            const int* __restrict__ batch,
                            float* __restrict__ pool, int n) {
    int gid = blockIdx.x * blockDim.x + threadIdx.x;
    int node = gid >> 5, lane = gid & 31;
    if (node >= n) return;
    int b = batch[node];
    float4 v = ((const float4*)(h + (size_t)node * HDIM))[lane];
    float* pp = pool + (size_t)b * HDIM + lane * 4;
    atomicAdd(pp + 0, v.x);
    atomicAdd(pp + 1, v.y);
    atomicAdd(pp + 2, v.z);
    atomicAdd(pp + 3, v.w);
}

// ---------------------------------------------------------------------
// WMMA fp32 GEMM, LDS-staged weight strip.
//
//   val = (acc + bias[n]) * (scale_const * (scale_vec ? scale_vec[n] : 1)) + (bias2 ? bias2[n] : 0)
//   if (relu) val = max(val, 0)
//   Y[m,n] = val;   if (Y2) Y2[m,n] = c2p[0] * val   (side write, e.g. fW0*z)
//
// The whole block shares one 16-wide N strip: the B tile (KIN x 16 f32,
// 8-16KB of the WGP's 320KB LDS) is staged ONCE per block, pre-swizzled
// into WMMA lane order so each lane's per-WMMA B pair is a single
// conflict-free 8-byte LDS read at (j*64 + lane*2) floats.
// Each of the 4 waves then sweeps TMW consecutive 16x16 M-tiles; the
// inner loop per WMMA is one global_load_b64 (A pair) + one LDS pair +
// one v_wmma_f32_16x16x4_f32. launch_bounds(128,1) gives the register
// allocator headroom so nothing spills to scratch. Out-of-range rows /
// cols use clamped loads (their products only feed never-stored
// outputs), so EXEC is all-ones at every WMMA as the ISA requires.
template<int KIN, int TMW>
__global__ __launch_bounds__(128, 1)
void gemm_wmma_kernel(const float* __restrict__ X, int lda,
                      const float* __restrict__ W, int ldb,
                      const float* __restrict__ bias,
                      const float* __restrict__ scale_vec, float scale_const,
                      const float* __restrict__ bias2,
                      int relu,
                      float* __restrict__ Y, int ldy,
                      float* __restrict__ Y2, int ldy2,
                      const float* __restrict__ c2p,
                      int M, int Nout) {
    __shared__ float ldsB[KIN * 16];

    const int tid  = threadIdx.x;
    const int lane = tid & 31;
    const int wave = tid >> 5;
    const int tN = (Nout + 15) >> 4;
    const int tM = (M + 15) >> 4;
    const int nt  = blockIdx.x % tN;           // N strip for the whole block
    const int mcb = blockIdx.x / tN;           // chunk of 4*TMW M-tiles
    const int n0  = nt << 4;

    // ---- cooperative stage of B tile into LDS (WMMA lane order) ----
    {
        const int c = tid & 15;
        int ncol = n0 + c; if (ncol > Nout - 1) ncol = Nout - 1;   // clamp
        for (int k = tid >> 4; k < KIN; k += 8) {
            // element (k, c) feeds lane ((k>>1)&1)*16 + c, pair j = k>>2, slot k&1
            const int off = (k >> 2) * 64 + ((k >> 1) & 1) * 32 + c * 2 + (k & 1);
            ldsB[off] = W[(size_t)k * ldb + ncol];
        }
    }
    __syncthreads();

    const int col  = lane & 15;
    const int half = lane >> 4;                // 0 or 1 (selects K pair)
    const int bcol = n0 + col;
    int bcol_c = bcol; if (bcol_c > Nout - 1) bcol_c = Nout - 1;   // clamp

    const float bs  = bias[bcol_c];
    const float sc  = scale_const * (scale_vec ? scale_vec[bcol_c] : 1.0f);
    const float b2v = bias2 ? bias2[bcol_c] : 0.0f;
    const float c2  = c2p ? c2p[0] : 0.0f;
    const int rowoff = half * 8;               // C/D layout: VGPR v -> M = v (+8 for lanes 16-31)
    const float* lb = ldsB + lane * 2;         // per-lane B base; pair j at +j*64 floats

    // ---- sweep TMW M-tiles reusing the LDS-resident B strip ----
    #pragma unroll 1
    for (int i = 0; i < TMW; ++i) {
        const int mt = (mcb * 4 + wave) * TMW + i;   // wave-uniform
        if (mt >= tM) break;
        const int m0 = mt << 4;
        int arow = m0 + col; if (arow > M - 1) arow = M - 1;       // clamp
        const float* xr = X + (size_t)arow * lda + 2 * half;

        v8f acc = {};
        #pragma unroll
        for (int j = 0; j < KIN / 4; ++j) {
            v2f a, b;
            a.x = xr[4 * j];
            a.y = xr[4 * j + 1];
            b.x = lb[j * 64];
            b.y = lb[j * 64 + 1];
            acc = __builtin_amdgcn_wmma_f32_16x16x4_f32(
                      false, a, false, b, (short)0, acc, false, false);
        }

        if (bcol < Nout) {
            #pragma unroll
            for (int v = 0; v < 8; ++v) {
                const int m = m0 + rowoff + v;
                if (m < M) {
                    float val = (acc[v] + bs) * sc + b2v;
                    if (relu) val = fmaxf(val, 0.0f);
                    Y[(size_t)m * ldy + bcol] = val;
                    if (Y2) Y2[(size_t)m * ldy2 + bcol] = c2 * val;
                }
            }
        }
    }
}

// ---------------------------------------------------------------------
#define TMW_TILES 8

static inline void launch_gemm(hipStream_t stream,
                               const float* X, int lda, const float* W, int ldb,
                               const float* bias, const float* scale_vec, float scale_const,
                               const float* bias2, int relu,
                               float* Y, int ldy, float* Y2, int ldy2, const float* c2p,
                               int M, int Kin, int Nout) {
    int tM = (M + 15) / 16, tN = (Nout + 15) / 16;
    int tilesPerBlock = 4 * TMW_TILES;         // 4 waves x TMW tiles
    int mchunks = (tM + tilesPerBlock - 1) / tilesPerBlock;
    int blocks = tN * mchunks;
    if (Kin == 256) {
        gemm_wmma_kernel<256, TMW_TILES><<<blocks, 128, 0, stream>>>(
            X, lda, W, ldb, bias, scale_vec, scale_const, bias2, relu,
            Y, ldy, Y2, ldy2, c2p, M, Nout);
    } else {
        gemm_wmma_kernel<128, TMW_TILES><<<blocks, 128, 0, stream>>>(
            X, lda, W, ldb, bias, scale_vec, scale_const, bias2, relu,
            Y, ldy, Y2, ldy2, c2p, M, Nout);
    }
}

static inline void launch_fill(hipStream_t stream, float* p, size_t nfloats) {
    int n4 = (int)(nfloats / 4);
    fill_zero_kernel<<<(n4 + 255) / 256, 256, 0, stream>>>(p, n4);
}

extern "C" void kernel_launch(void* const* d_in, const int* in_sizes, int n_in,
                              void* d_out, int out_size, void* d_ws, size_t ws_size,
                              hipStream_t stream) {
    (void)in_sizes; (void)n_in; (void)out_size; (void)ws_size;

    const float* x           = (const float*)d_in[0];
    const int*   edge_index  = (const int*)  d_in[1];   // [O,2,E]
    const float* edge_weight = (const float*)d_in[2];   // [O,E]
    const int*   batch       = (const int*)  d_in[3];   // [N]
    const float* W_in  = (const float*)d_in[4];         // [L,O,F,H]
    const float* b_in  = (const float*)d_in[5];         // [L,O,H]
    const float* fW    = (const float*)d_in[6];         // [L,O,K+1]
    const float* W_out = (const float*)d_in[7];         // [L,2H,H]
    const float* b_out = (const float*)d_in[8];         // [L,H]
    const float* Wn1   = (const float*)d_in[9];
    const float* bn1   = (const float*)d_in[10];
    const float* g1    = (const float*)d_in[11];
    const float* be1   = (const float*)d_in[12];
    const float* Wn2   = (const float*)d_in[13];
    const float* bn2   = (const float*)d_in[14];
    const float* g2    = (const float*)d_in[15];
    const float* be2   = (const float*)d_in[16];
    const float* W1    = (const float*)d_in[17];        // [H,H]
    const float* b1    = (const float*)d_in[18];        // [H]
    const float* W2    = (const float*)d_in[19];        // [H,C]
    const float* b2    = (const float*)d_in[20];        // [C]
    float* out = (float*)d_out;                         // [G,C]

    // workspace layout (floats)
    const size_t NH = (size_t)N_NODES * HDIM;
    float* ws    = (float*)d_ws;
    float* bufH  = ws;                  // [N,H]   layer output h
    float* xcat  = bufH + NH;           // [N,2H]  concat(hid_o0, hid_o1)
    float* zA    = xcat + 2 * NH;       // [N,H]   spmm ping
    float* zB    = zA + NH;             // [N,H]   spmm pong
    float* tmp   = zB + NH;             // [N,H]   xc
    float* pool  = tmp + NH;            // [G,H]
    float* p1    = pool + (size_t)GRAPHS * HDIM;  // [G,H]

    const float invs = 1.0f / sqrtf(1.0f + 1e-5f);  // eval-mode BN: g / sqrt(1+eps)

    const int spmm_threads = EDGES * 32;
    const int spmm_blocks  = (spmm_threads + 255) / 256;
    const int nh_threads   = N_NODES * 32;
    const int nh_blocks    = (nh_threads + 255) / 256;

    const float* hcur = x;
    for (int l = 0; l < LAYERS; ++l) {
        for (int o = 0; o < ORDER; ++o) {
            const float* Wi   = W_in + (size_t)(l * ORDER + o) * FDIM * HDIM;
            const float* bi   = b_in + (size_t)(l * ORDER + o) * HDIM;
            const float* fWlo = fW   + (size_t)(l * ORDER + o) * (HOPS + 1);

            // z = h @ W_in + b   (into zA), side-write  hid = fW[0]*z  into xcat column block o
            launch_gemm(stream, hcur, HDIM, Wi, HDIM, bi, nullptr, 1.0f, nullptr, 0,
                        zA, HDIM, xcat + o * HDIM, 2 * HDIM, fWlo,
                        N_NODES, FDIM, HDIM);

            const int*   dstp = edge_index + (size_t)o * 2 * EDGES;
            const int*   srcp = dstp + EDGES;
            const float* wp   = edge_weight + (size_t)o * EDGES;

            float* zin = zA; float* zout = zB;
            for (int k = 0; k < HOPS; ++k) {
                launch_fill(stream, zout, NH);
                spmm_atomic_kernel<<<spmm_blocks, 256, 0, stream>>>(dstp, srcp, wp,
                                                                    zin, zout, EDGES);
                axpy_strided_kernel<<<nh_blocks, 256, 0, stream>>>(xcat + o * HDIM, 2 * HDIM,
                                                                   zout, fWlo + k + 1, N_NODES);
                float* t = zin; zin = zout; zout = t;
            }
        }

        // xc = concat @ W_out + b_out
        launch_gemm(stream, xcat, 2 * HDIM, W_out + (size_t)l * 2 * HDIM * HDIM, HDIM,
                    b_out + (size_t)l * HDIM, nullptr, 1.0f, nullptr, 0,
                    tmp, HDIM, nullptr, 0, nullptr, N_NODES, 2 * HDIM, HDIM);
        // t = relu(bn(xc @ Wn1 + bn1))
        launch_gemm(stream, tmp, HDIM, Wn1 + (size_t)l * HDIM * HDIM, HDIM,
                    bn1 + (size_t)l * HDIM, g1 + (size_t)l * HDIM, invs,
                    be1 + (size_t)l * HDIM, 1,
                    zA, HDIM, nullptr, 0, nullptr, N_NODES, HDIM, HDIM);
        // h = relu(bn(t @ Wn2 + bn2))
        launch_gemm(stream, zA, HDIM, Wn2 + (size_t)l * HDIM * HDIM, HDIM,
                    bn2 + (size_t)l * HDIM, g2 + (size_t)l * HDIM, invs,
                    be2 + (size_t)l * HDIM, 1,
                    bufH, HDIM, nullptr, 0, nullptr, N_NODES, HDIM, HDIM);
        hcur = bufH;
    }

    // global_add_pool
    launch_fill(stream, pool, (size_t)GRAPHS * HDIM);
    pool_kernel<<<nh_blocks, 256, 0, stream>>>(hcur, batch, pool, N_NODES);

    // p1 = relu(pool @ W1 + b1)
    launch_gemm(stream, pool, HDIM, W1, HDIM, b1, nullptr, 1.0f, nullptr, 1,
                p1, HDIM, nullptr, 0, nullptr, GRAPHS, HDIM, HDIM);
    // out = p1 @ W2 + b2   (N=10, store-masked)
    launch_gemm(stream, p1, HDIM, W2, CDIM, b2, nullptr, 1.0f, nullptr, 0,
                out, CDIM, nullptr, 0, nullptr, GRAPHS, HDIM, CDIM);
}